// LogNCDE_9225589752316
// MI455X (gfx1250) — compile-verified
//
#include <hip/hip_runtime.h>
#include <math.h>

#define WIN  16
#define DD   6
#define SS   128
#define HH   128
#define NW   32      // number of signature windows = (513-1)/16
#define TT   513
#define OUTD 10
#define NTHREADS 256
#define NWAVES   8
#define WPAD 130     // padded LDS row stride for cached weights (bank-conflict-free GEMV)

typedef __attribute__((ext_vector_type(2))) float v2f;
typedef __attribute__((ext_vector_type(8))) float v8f;

__device__ __forceinline__ float softplus_f(float x) {
  return fmaxf(x, 0.0f) + log1pf(expf(-fabsf(x)));   // numerically stable
}
__device__ __forceinline__ float sigmoid_f(float x) {
  return 1.0f / (1.0f + expf(-x));
}

// Yout = (optional diag(scale)) * ( W[Mout x 128] @ X[128 x 16] ), fp32 WMMA 16x16x4.
// Tiles are distributed over the 8 waves. WST = row stride of W (130 for LDS-cached,
// 128 for global). Fragment layouts per CDNA5 ISA:
//  A (16x4 f32): lanes 0-15: v0=K+0, v1=K+1 ; lanes 16-31: v0=K+2, v1=K+3 (M = lane&15)
//  B (4x16 f32): same K striping, N = lane&15
//  C/D (16x16 f32): VGPR r -> M = r (+8 for lanes>=16), N = lane&15
template <int WST, bool SCALE>
__device__ __forceinline__ void mm_layer(const float* __restrict__ Wsrc,
                                         const float (*__restrict__ Xin)[16],
                                         float (*__restrict__ Yout)[16],
                                         const float* __restrict__ scale,
                                         int mtiles, int wave, int lane) {
  const int l15  = lane & 15;
  const int half = lane >> 4;      // 0 or 1
  const int kc0  = half << 1;      // 0 or 2
  for (int tile = wave; tile < mtiles; tile += NWAVES) {
    const int m0 = tile << 4;
    const float* wr = Wsrc + (m0 + l15) * WST;
    v8f c = {};
    #pragma unroll 8
    for (int kt = 0; kt < 32; ++kt) {
      const int k = (kt << 2) + kc0;
      v2f a, bfr;
      a[0]   = wr[k];
      a[1]   = wr[k + 1];
      bfr[0] = Xin[k][l15];
      bfr[1] = Xin[k + 1][l15];
      c = __builtin_amdgcn_wmma_f32_16x16x4_f32(
          /*neg_a=*/false, a, /*neg_b=*/false, bfr,
          /*c_mod=*/(short)0, c, /*reuse_a=*/false, /*reuse_b=*/false);
    }
    const int mo = m0 + (half << 3);
    #pragma unroll
    for (int r = 0; r < 8; ++r) {
      const int m = mo + r;
      Yout[m][l15] = SCALE ? (scale[m] * c[r]) : c[r];
    }
  }
}

__global__ void __launch_bounds__(NTHREADS, 1)
logncde_kernel(const float* __restrict__ x,
               const float* __restrict__ Wi0, const float* __restrict__ bi0,
               const float* __restrict__ Wi1, const float* __restrict__ bi1,
               const float* __restrict__ Wi2, const float* __restrict__ bi2,
               const float* __restrict__ Wv0, const float* __restrict__ bv0,
               const float* __restrict__ Wv1, const float* __restrict__ bv1,
               const float* __restrict__ Wv2, const float* __restrict__ bv2,
               const float* __restrict__ Wr,  const float* __restrict__ br,
               float* __restrict__ out) {
  __shared__ __align__(16) float W0s[HH * WPAD];    // Wv0 cached in LDS (padded rows)
  __shared__ __align__(16) float W1s[HH * WPAD];    // Wv1 cached in LDS
  __shared__ __align__(16) float sh_h [SS];
  __shared__ __align__(16) float sh_a1[HH];         // z1 = softplus(pre0)
  __shared__ __align__(16) float sh_s0[HH];         // sigmoid(pre0)
  __shared__ __align__(16) float sh_a2[HH];         // z2 = softplus(pre1)
  __shared__ __align__(16) float sh_s1[HH];         // sigmoid(pre1)
  __shared__ __align__(16) float sh_V [DD * SS];    // tanh(pre2), row-major [D][S]
  __shared__ __align__(16) float sh_X [2][HH][16];  // ping-pong WMMA RHS (cols>=6 stay 0)
  __shared__ __align__(16) float sh_Y [DD * SS][16];// WMMA output staging (T2)
  __shared__ float sh_l1[NW][DD];
  __shared__ float sh_ar[NW][DD * DD];              // full antisymmetric Levy areas
  __shared__ float sh_x0[DD];

  const int b    = blockIdx.x;
  const int tid  = threadIdx.x;
  const int lane = tid & 31;
  const int wave = tid >> 5;
  const float* xb = x + b * TT * DD;

  // One-time: cache Wv0/Wv1 into padded LDS rows (float2 granularity keeps 8B align).
  for (int e = tid; e < HH * (HH / 2); e += NTHREADS) {
    const int r = e >> 6;
    const int c = (e & 63) << 1;
    *(float2*)&W0s[r * WPAD + c] = *(const float2*)&Wv0[r * HH + c];
    *(float2*)&W1s[r * WPAD + c] = *(const float2*)&Wv1[r * HH + c];
  }
  // Zero both X buffers once; cols 6..15 remain exactly zero forever after.
  for (int e = tid; e < 2 * HH * 16; e += NTHREADS) (&sh_X[0][0][0])[e] = 0.0f;

  // Pull Wv2 into L2 early (global_prefetch); it stays resident all 32 steps.
  for (int line = tid; line < (DD * SS * HH) / 32; line += NTHREADS)
    __builtin_prefetch(Wv2 + line * 32, 0, 3);

  // ---- windowed depth-2 log-signature: one thread per window ----
  if (tid < NW) {
    float cum[DD], l1[DD], cr[DD][DD];
    #pragma unroll
    for (int i = 0; i < DD; ++i) {
      cum[i] = 0.0f; l1[i] = 0.0f;
      #pragma unroll
      for (int j = 0; j < DD; ++j) cr[i][j] = 0.0f;
    }
    const float* p = xb + tid * WIN * DD;
    for (int k = 0; k < WIN; ++k) {
      float dx[DD];
      #pragma unroll
      for (int d = 0; d < DD; ++d) dx[d] = p[(k + 1) * DD + d] - p[k * DD + d];
      #pragma unroll
      for (int i = 0; i < DD; ++i)
        #pragma unroll
        for (int j = 0; j < DD; ++j) cr[i][j] += cum[i] * dx[j];
      #pragma unroll
      for (int d = 0; d < DD; ++d) { l1[d] += dx[d]; cum[d] += dx[d]; }
    }
    #pragma unroll
    for (int d = 0; d < DD; ++d) sh_l1[tid][d] = l1[d];
    #pragma unroll
    for (int i = 0; i < DD; ++i)
      #pragma unroll
      for (int j = 0; j < DD; ++j)
        sh_ar[tid][i * DD + j] = 0.5f * (cr[i][j] - cr[j][i]);
  }
  if (tid < DD) sh_x0[tid] = xb[tid];
  __syncthreads();

  // ---- h0 = init MLP(x[0]) ----
  if (tid < HH) {
    float acc = bi0[tid];
    #pragma unroll
    for (int d = 0; d < DD; ++d) acc += Wi0[tid * DD + d] * sh_x0[d];
    sh_a1[tid] = softplus_f(acc);
  }
  __syncthreads();
  if (tid < HH) {
    const float* wr = Wi1 + tid * HH;
    float acc = bi1[tid];
    #pragma unroll 8
    for (int k = 0; k < HH; ++k) acc += wr[k] * sh_a1[k];
    sh_a2[tid] = softplus_f(acc);
  }
  __syncthreads();
  if (tid < SS) {
    const float* wr = Wi2 + tid * HH;
    float acc = bi2[tid];
    #pragma unroll 8
    for (int k = 0; k < HH; ++k) acc += wr[k] * sh_a2[k];
    sh_h[tid] = acc;
  }
  __syncthreads();
  if (tid < OUTD) {                       // readout of h0 (row 0)
    const float* wr = Wr + tid * SS;
    float acc = br[tid];
    #pragma unroll 8
    for (int k = 0; k < SS; ++k) acc += wr[k] * sh_h[k];
    out[b * (NW + 1) * OUTD + tid] = acc;
  }
  __syncthreads();

  // ---- log-ODE scan: 32 sequential steps ----
  for (int t = 0; t < NW; ++t) {
    // forward layer 0 (Wv0 from LDS): pre0 = Wv0 @ h + b
    if (tid < HH) {
      const float* wr = &W0s[tid * WPAD];
      float acc = bv0[tid];
      #pragma unroll 8
      for (int k = 0; k < SS; ++k) acc += wr[k] * sh_h[k];
      sh_a1[tid] = softplus_f(acc);
      sh_s0[tid] = sigmoid_f(acc);
    }
    __syncthreads();
    // forward layer 1 (Wv1 from LDS)
    if (tid < HH) {
      const float* wr = &W1s[tid * WPAD];
      float acc = bv1[tid];
      #pragma unroll 8
      for (int k = 0; k < HH; ++k) acc += wr[k] * sh_a1[k];
      sh_a2[tid] = softplus_f(acc);
      sh_s1[tid] = sigmoid_f(acc);
    }
    __syncthreads();
    // forward layer 2 (Wv2 from global/L2): V = tanh(pre2); also scatter V^T into X0
    for (int r = tid; r < DD * SS; r += NTHREADS) {
      const float* wr = Wv2 + r * HH;
      float acc = bv2[r];
      #pragma unroll 8
      for (int k = 0; k < HH; ++k) acc += wr[k] * sh_a2[k];
      const float v = tanhf(acc);
      sh_V[r] = v;
      sh_X[0][r & (SS - 1)][r >> 7] = v;   // X0[k][i] = V[i][k]; cols 6..15 already 0
    }
    __syncthreads();
    // T0 = Wv0 @ X0 ; U = diag(s0)*T0 -> X1        (8 tiles, fused epilogue)
    mm_layer<WPAD, true>(W0s, sh_X[0], sh_X[1], sh_s0, SS / 16, wave, lane);
    __syncthreads();
    // T1 = Wv1 @ X1 ; Wt = diag(s1)*T1 -> X0       (8 tiles, fused epilogue)
    mm_layer<WPAD, true>(W1s, sh_X[1], sh_X[0], sh_s1, SS / 16, wave, lane);
    __syncthreads();
    // T2 = Wv2 @ X0 -> Y                           (48 tiles)
    mm_layer<HH, false>(Wv2, sh_X[0], sh_Y, nullptr, (DD * SS) / 16, wave, lane);
    __syncthreads();
    // combine: h += V^T l1 + sum_{i,j} area[i,j] * (1 - V[j,s]^2) * T2[j*S+s][i]
    if (tid < SS) {
      float hv = sh_h[tid];
      #pragma unroll
      for (int i = 0; i < DD; ++i) hv += sh_l1[t][i] * sh_V[i * SS + tid];
      #pragma unroll
      for (int j = 0; j < DD; ++j) {
        const float vj = sh_V[j * SS + tid];
        const float g  = 1.0f - vj * vj;
        float acc = 0.0f;
        #pragma unroll
        for (int i = 0; i < DD; ++i) acc += sh_ar[t][i * DD + j] * sh_Y[j * SS + tid][i];
        hv += g * acc;
      }
      sh_h[tid] = hv;   // each thread touches only its own element
    }
    __syncthreads();
    // evolving readout
    if (tid < OUTD) {
      const float* wr = Wr + tid * SS;
      float acc = br[tid];
      #pragma unroll 8
      for (int k = 0; k < SS; ++k) acc += wr[k] * sh_h[k];
      out[b * (NW + 1) * OUTD + (t + 1) * OUTD + tid] = acc;
    }
    __syncthreads();
  }
}

extern "C" void kernel_launch(void* const* d_in, const int* in_sizes, int n_in,
                              void* d_out, int out_size, void* d_ws, size_t ws_size,
                              hipStream_t stream) {
  const float* x   = (const float*)d_in[0];
  const float* Wi0 = (const float*)d_in[1];
  const float* bi0 = (const float*)d_in[2];
  const float* Wi1 = (const float*)d_in[3];
  const float* bi1 = (const float*)d_in[4];
  const float* Wi2 = (const float*)d_in[5];
  const float* bi2 = (const float*)d_in[6];
  const float* Wv0 = (const float*)d_in[7];
  const float* bv0 = (const float*)d_in[8];
  const float* Wv1 = (const float*)d_in[9];
  const float* bv1 = (const float*)d_in[10];
  const float* Wv2 = (const float*)d_in[11];
  const float* bv2 = (const float*)d_in[12];
  const float* Wr  = (const float*)d_in[13];
  const float* br  = (const float*)d_in[14];
  float* out = (float*)d_out;

  const int B = in_sizes[0] / (TT * DD);   // 8
  logncde_kernel<<<dim3(B), dim3(NTHREADS), 0, stream>>>(
      x, Wi0, bi0, Wi1, bi1, Wi2, bi2,
      Wv0, bv0, Wv1, bv1, Wv2, bv2, Wr, br, out);
}